// StructureRerankModel_3607772528969
// MI455X (gfx1250) — compile-verified
//
#include <hip/hip_runtime.h>
#include <hip/hip_bf16.h>

// ---------------- problem constants (match reference) ----------------
#define VOCAB     30523
#define SPAN_TOK  30522
#define DIM       768
#define N_DOCS    64
#define N_PASS    1024
#define TITLE_LEN 128
#define SEQ       256
#define N_EDGES   32768
#define N_NODES   (N_DOCS + 2 * N_PASS)   // 2112
#define N_REL     2
#define TAU       0.2f

#define APAD      4
#define ASTRIDE   (DIM + APAD)            // 772 ≡ 4 (mod 64) -> conflict-free ds_load_b64

typedef __attribute__((ext_vector_type(2))) float v2f;
typedef __attribute__((ext_vector_type(8))) float v8f;

// ---------------------------------------------------------------------
// 1) Gather node features: x0[node][0..767]
// ---------------------------------------------------------------------
__global__ __launch_bounds__(256) void build_x_kernel(
    const int* __restrict__ doc_ids, const int* __restrict__ pass_ids,
    const float* __restrict__ embed, float* __restrict__ x0)
{
    const int node = blockIdx.x;
    const int tid  = threadIdx.x;
    __shared__ int s_tok;
    if (tid == 0) s_tok = 0;
    __syncthreads();

    if (node < N_DOCS) {
        if (tid == 0) s_tok = doc_ids[node * TITLE_LEN];
    } else if (node < N_DOCS + N_PASS) {
        if (tid == 0) s_tok = pass_ids[(node - N_DOCS) * SEQ];
    } else {
        const int p = node - (N_DOCS + N_PASS);
        const int t = pass_ids[p * SEQ + tid];      // blockDim == SEQ == 256
        if (t == SPAN_TOK) s_tok = t;               // exactly one match per row
    }
    __syncthreads();

    const float* src = embed + (size_t)s_tok * DIM;
    float*       dst = x0    + (size_t)node  * DIM;
    for (int c = tid; c < DIM; c += 256) dst[c] = src[c];
}

// ---------------------------------------------------------------------
// 2) Deterministic, atomic-free mean aggregation (ballot edge scan).
//    agg[r][node][c] = (sum over matching src rows, fixed order) / max(cnt,1)
// ---------------------------------------------------------------------
__global__ __launch_bounds__(256) void aggregate_mean_kernel(
    const float* __restrict__ x,    const int* __restrict__ esrc,
    const int* __restrict__ edst,   const int* __restrict__ etype,
    float* __restrict__ agg)        // [N_REL][N_NODES][DIM]
{
    const int node = blockIdx.x;
    const int tid  = threadIdx.x;
    const int wid  = tid >> 5;          // 0..7
    const int lane = tid & 31;

    float acc0[DIM / 32], acc1[DIM / 32];   // 24 + 24 per-lane accumulators
#pragma unroll
    for (int j = 0; j < DIM / 32; ++j) { acc0[j] = 0.f; acc1[j] = 0.f; }
    int cnt0 = 0, cnt1 = 0;

    const int CH = N_EDGES / 8;         // 4096 edges per wave
    const int e0 = wid * CH;
    for (int eb = e0; eb < e0 + CH; eb += 32) {
        const int e = eb + lane;
        const int d = edst[e];                       // coalesced b32
        const bool hit = (d == node);
        int my_src = 0, my_r = -1;
        if (hit) { my_src = esrc[e]; my_r = etype[e]; }
        unsigned long long mask = __ballot(hit);
        while (mask) {
            const int b = __ffsll(mask) - 1;         // smallest e first
            mask &= mask - 1;
            const int srcn = __shfl(my_src, b, 32);
            const int r    = __shfl(my_r,  b, 32);
            const float* xs = x + (size_t)srcn * DIM;
            if (r == 0) {
                ++cnt0;
#pragma unroll
                for (int j = 0; j < DIM / 32; ++j) acc0[j] += xs[lane + 32 * j];
            } else if (r == 1) {
                ++cnt1;
#pragma unroll
                for (int j = 0; j < DIM / 32; ++j) acc1[j] += xs[lane + 32 * j];
            }
        }
    }

    __shared__ float s_acc[8][N_REL][DIM];   // 49 KB
    __shared__ int   s_cnt[8][N_REL];
#pragma unroll
    for (int j = 0; j < DIM / 32; ++j) {
        s_acc[wid][0][lane + 32 * j] = acc0[j];
        s_acc[wid][1][lane + 32 * j] = acc1[j];
    }
    if (lane == 0) { s_cnt[wid][0] = cnt0; s_cnt[wid][1] = cnt1; }
    __syncthreads();

    int ctot[N_REL];
#pragma unroll
    for (int r = 0; r < N_REL; ++r) {
        int n = 0;
#pragma unroll
        for (int w = 0; w < 8; ++w) n += s_cnt[w][r];
        ctot[r] = n;
    }

    for (int idx = tid; idx < N_REL * DIM; idx += 256) {   // 6 per thread
        const int r = idx / DIM, c = idx % DIM;
        float t = 0.f;
#pragma unroll
        for (int w = 0; w < 8; ++w) t += s_acc[w][r][c];   // fixed order
        agg[((size_t)r * N_NODES + node) * DIM + c] =
            t / fmaxf((float)ctot[r], 1.0f);
    }
}

// ---------------------------------------------------------------------
// 3) Fused RGCN GEMM with V_WMMA_F32_16X16X4_F32.
//    Helpers are force-inlined and called once per source matrix with the
//    kernel-arg pointer used DIRECTLY (no pointer-array select), so
//    infer-address-spaces promotes B loads to global_load_b32 (LOADcnt
//    only) instead of flat_load (which would also tie up DScnt / the LDS
//    path that the ds_load_2addr_b64 A-fragments use).
// ---------------------------------------------------------------------
__device__ __forceinline__ void stage_A_tile(
    float* sA, const float* __restrict__ A, int row0, int tid)
{
    for (int i = tid; i < 16 * DIM; i += 256) {      // 48 floats/thread
        const int rr = i / DIM, cc = i - rr * DIM;
        sA[rr * ASTRIDE + cc] = A[(size_t)(row0 + rr) * DIM + cc];
    }
}

__device__ __forceinline__ void mm_tile(
    v8f& c, const float* aptr /* LDS */, const float* __restrict__ B,
    int col, int half)
{
    const float* bptr = B + col + (size_t)(2 * half) * DIM;   // global
#pragma unroll 8
    for (int k = 0; k < DIM; k += 4) {
        v2f a = { aptr[k], aptr[k + 1] };                     // ds_load_b64
        v2f b = { bptr[(size_t)k * DIM], bptr[(size_t)k * DIM + DIM] };
        c = __builtin_amdgcn_wmma_f32_16x16x4_f32(
                false, a, false, b, (short)0, c, false, false);
    }
}

__global__ __launch_bounds__(256) void rgcn_gemm3_kernel(
    const float* __restrict__ A0, const float* __restrict__ A1,
    const float* __restrict__ A2,
    const float* __restrict__ B0, const float* __restrict__ B1,
    const float* __restrict__ B2,
    const float* __restrict__ bias, float* __restrict__ out)
{
    __shared__ float sA[16 * ASTRIDE];                   // 49.4 KB

    const int NT   = DIM / 16;                           // 48 col tiles
    const int tid  = threadIdx.x;
    const int wid  = tid >> 5;                           // 0..7
    const int lane = tid & 31;
    const int m    = lane & 15;
    const int half = lane >> 4;

    const int tm   = blockIdx.x / (NT / 8);              // row tile 0..131
    const int tg   = blockIdx.x % (NT / 8);              // col group 0..5
    const int tn   = tg * 8 + wid;
    const int row0 = tm * 16;
    const int col  = tn * 16 + m;

    v8f c = {0.f, 0.f, 0.f, 0.f, 0.f, 0.f, 0.f, 0.f};
    const float* aF = &sA[m * ASTRIDE + 2 * half];       // per-lane A frag base

    stage_A_tile(sA, A0, row0, tid);
    __syncthreads();
    mm_tile(c, aF, B0, col, half);
    __syncthreads();

    stage_A_tile(sA, A1, row0, tid);
    __syncthreads();
    mm_tile(c, aF, B1, col, half);
    __syncthreads();

    stage_A_tile(sA, A2, row0, tid);
    __syncthreads();
    mm_tile(c, aF, B2, col, half);

    const float bv = bias[col];
#pragma unroll
    for (int v = 0; v < 8; ++v) {
        const int orow = row0 + v + 8 * half;
        out[(size_t)orow * DIM + col] = c[v] + bv;
    }
}

// ---------------------------------------------------------------------
// 4) Head: out[p] = exp(sigmoid(x2[span_p] . clf_w + clf_b) / TAU)
// ---------------------------------------------------------------------
__global__ __launch_bounds__(256) void classify_kernel(
    const float* __restrict__ x2, const float* __restrict__ clf_w,
    const float* __restrict__ clf_b, float* __restrict__ out)
{
    const int p = blockIdx.x;
    const float* row = x2 + (size_t)(N_DOCS + N_PASS + p) * DIM;

    float partial = 0.f;
    for (int c = threadIdx.x; c < DIM; c += 256)
        partial += row[c] * clf_w[c];

#pragma unroll
    for (int off = 16; off > 0; off >>= 1)
        partial += __shfl_down(partial, off, 32);

    __shared__ float sred[8];
    if ((threadIdx.x & 31) == 0) sred[threadIdx.x >> 5] = partial;
    __syncthreads();
    if (threadIdx.x == 0) {
        float t = 0.f;
#pragma unroll
        for (int w = 0; w < 8; ++w) t += sred[w];
        const float sc = 1.f / (1.f + __expf(-(t + clf_b[0])));
        out[p] = __expf(sc / TAU);
    }
}

// ---------------------------------------------------------------------
extern "C" void kernel_launch(void* const* d_in, const int* in_sizes, int n_in,
                              void* d_out, int out_size, void* d_ws, size_t ws_size,
                              hipStream_t stream)
{
    (void)in_sizes; (void)n_in; (void)out_size; (void)ws_size;

    const int*   doc_ids    = (const int*)  d_in[0];
    const int*   pass_ids   = (const int*)  d_in[1];
    const int*   edge_index = (const int*)  d_in[2];
    const int*   edge_type  = (const int*)  d_in[3];
    const float* embed      = (const float*)d_in[4];
    const float* w_root1    = (const float*)d_in[5];
    const float* w_rel1     = (const float*)d_in[6];
    const float* b1         = (const float*)d_in[7];
    const float* w_root2    = (const float*)d_in[8];
    const float* w_rel2     = (const float*)d_in[9];
    const float* b2         = (const float*)d_in[10];
    const float* clf_w      = (const float*)d_in[11];
    const float* clf_b      = (const float*)d_in[12];
    float*       out        = (float*)d_out;

    // workspace layout (floats): x0 | x1 | x2 | agg[2]
    const size_t NND = (size_t)N_NODES * DIM;            // 1,622,016
    float* ws  = (float*)d_ws;
    float* x0  = ws;
    float* x1  = x0 + NND;
    float* x2  = x1 + NND;
    float* agg = x2 + NND;                               // 2*NND

    const int* esrc = edge_index;                        // edge_index[0]
    const int* edst = edge_index + N_EDGES;              // edge_index[1]

    const int GEMM_BLOCKS = (N_NODES / 16) * (DIM / 16 / 8);   // 132*6 = 792

    build_x_kernel<<<N_NODES, 256, 0, stream>>>(doc_ids, pass_ids, embed, x0);

    // ---- RGCN layer 1 ----
    aggregate_mean_kernel<<<N_NODES, 256, 0, stream>>>(x0, esrc, edst, edge_type, agg);
    rgcn_gemm3_kernel<<<GEMM_BLOCKS, 256, 0, stream>>>(
        x0, agg, agg + NND,
        w_root1, w_rel1, w_rel1 + (size_t)DIM * DIM, b1, x1);

    // ---- RGCN layer 2 ----
    aggregate_mean_kernel<<<N_NODES, 256, 0, stream>>>(x1, esrc, edst, edge_type, agg);
    rgcn_gemm3_kernel<<<GEMM_BLOCKS, 256, 0, stream>>>(
        x1, agg, agg + NND,
        w_root2, w_rel2, w_rel2 + (size_t)DIM * DIM, b2, x2);

    // ---- head ----
    classify_kernel<<<N_PASS, 256, 0, stream>>>(x2, clf_w, clf_b, out);
}